// CorrMLP_13898514170206
// MI455X (gfx1250) — compile-verified
//
#include <hip/hip_runtime.h>
#include <cstddef>

#define B_    4
#define CF_   128
#define H_    160
#define W_    160
#define C_    81
#define P_    162
#define KPAD  84
#define HW_   (H_*W_)
#define NTOK_ (B_*HW_)

typedef __attribute__((ext_vector_type(2))) float v2f;
typedef __attribute__((ext_vector_type(8))) float v8f;

__device__ __forceinline__ float gelu_exact(float x) {
  return 0.5f * x * (1.0f + erff(x * 0.7071067811865476f));
}

// Stage B (global, Kact x Nact row-major, ld=ldb) into LDS transposed+padded:
// sBT[n*KPAD + k], n in [0,Npad), k in [0,KPAD), zero outside (Kact,Nact).
// Global reads are coalesced (consecutive threads -> consecutive n of one row k).
__device__ __forceinline__ void stage_BT(float* __restrict__ sBT,
                                         const float* __restrict__ gB, int ldb,
                                         int Kact, int Nact, int Npad) {
  for (int e = threadIdx.x; e < Npad * KPAD; e += 256) {
    const int n = e % Npad;
    const int k = e / Npad;
    float v = 0.f;
    if (n < Nact && k < Kact) v = gB[(size_t)k * ldb + n];
    sBT[n * KPAD + k] = v;
  }
}

// -------- wave-cooperative f32 WMMA GEMM, all-LDS operands.
// C(64 x Ntiles*16) = A(LDS 64 x KPAD) @ B^T(LDS Npad x KPAD), blockDim = 256 (8 waves).
// A frag (V_WMMA_F32_16X16X4_F32): lane L holds A[m=L&15][k=2*(L>>4)+v], v in {0,1}.
// B frag mirrors: lane L holds B[k=2*(L>>4)+v][n=L&15] == sBT[n][k...].
// C/D: VGPR r -> (M = r + 8*(L>>4), N = L&15) inside the 16x16 tile.
__device__ __forceinline__ void wmma_gemm_lds(const float* __restrict__ sA,
                                              const float* __restrict__ sBT,
                                              float* __restrict__ sC, int ldc,
                                              int Ntiles, int Nstore) {
  const int lane = threadIdx.x & 31;
  const int wave = __builtin_amdgcn_readfirstlane(threadIdx.x >> 5);
  const int m15  = lane & 15;
  const int kh   = lane >> 4;
  const int ntot = 4 * Ntiles;
  for (int tile = wave; tile < ntot; tile += 8) {
    const int mt = tile / Ntiles;
    const int nt = tile % Ntiles;
    const float* arow = sA  + (mt*16 + m15) * KPAD + 2*kh;
    const float* brow = sBT + (nt*16 + m15) * KPAD + 2*kh;
    v8f acc = {0.f,0.f,0.f,0.f,0.f,0.f,0.f,0.f};
    #pragma unroll
    for (int k0 = 0; k0 < KPAD; k0 += 4) {
      const v2f a = *(const v2f*)(arow + k0);   // 8B-aligned: KPAD*4 and 2*kh*4 are 8B multiples
      const v2f b = *(const v2f*)(brow + k0);
      acc = __builtin_amdgcn_wmma_f32_16x16x4_f32(false, a, false, b, (short)0, acc,
                                                  false, false);
    }
    const int row0 = mt*16 + kh*8;
    const int col  = nt*16 + m15;
    if (col < Nstore) {
      #pragma unroll
      for (int r = 0; r < 8; ++r) sC[(size_t)(row0 + r) * ldc + col] = acc[r];
    }
  }
}

// -------- correlation cost volume: (B,128,H,W)x2 -> (B,H,W,81), /(128*128)
__global__ __launch_bounds__(256)
void corr_kernel(const float* __restrict__ x1, const float* __restrict__ x2,
                 float* __restrict__ xin) {
  __shared__ float s1[16*16];
  __shared__ float s2[24*24];
  const int n  = blockIdx.y;
  const int ty = (blockIdx.x / 10) * 16;
  const int tx = (blockIdx.x % 10) * 16;
  const int tid = threadIdx.x;
  const int py = tid >> 4, px = tid & 15;
  const int h = ty + py, w = tx + px;
  float acc[81];
  #pragma unroll
  for (int d = 0; d < 81; ++d) acc[d] = 0.f;
  const size_t cs = (size_t)HW_;
  const float* p1 = x1 + (size_t)n * CF_ * cs;
  const float* p2 = x2 + (size_t)n * CF_ * cs;
  for (int c = 0; c < CF_; ++c) {
    s1[tid] = p1[c*cs + (size_t)h*W_ + w];
    for (int e = tid; e < 576; e += 256) {
      int hy = ty - 4 + e / 24;
      int hx = tx - 4 + e % 24;
      float v = 0.f;
      if (hy >= 0 && hy < H_ && hx >= 0 && hx < W_) v = p2[c*cs + (size_t)hy*W_ + hx];
      s2[e] = v;
    }
    __syncthreads();
    const float a = s1[tid];
    #pragma unroll
    for (int di = 0; di < 9; ++di) {
      #pragma unroll
      for (int dj = 0; dj < 9; ++dj)
        acc[di*9 + dj] = fmaf(a, s2[(py + di)*24 + (px + dj)], acc[di*9 + dj]);
    }
    __syncthreads();
  }
  float* o = xin + (((size_t)n*H_ + h)*W_ + w) * C_;
  const float sc = 1.f / (128.f * 128.f);
  #pragma unroll
  for (int d = 0; d < 81; ++d) o[d] = acc[d] * sc;
}

// -------- fused win_gmlp branch kernel: one block = WPB windows (<=64 tokens)
template<int F, int WPB>
__global__ __launch_bounds__(256)
void branch_kernel(const float* __restrict__ xin,
                   const float* __restrict__ mwg, const float* __restrict__ mwb,
                   const float* __restrict__ lng, const float* __restrict__ lnb,
                   const float* __restrict__ inW, const float* __restrict__ inb,
                   const float* __restrict__ slg, const float* __restrict__ slb,
                   const float* __restrict__ gW,  const float* __restrict__ gb,
                   const float* __restrict__ outW, const float* __restrict__ outb,
                   float* __restrict__ bout) {
  constexpr int F2 = F * F;
  constexpr int M  = WPB * F2;
  constexpr int GH = (H_ + F - 1) / F;
  constexpr int GWn = (W_ + F - 1) / F;
  constexpr int LDC = 162;
  const int nWin = B_ * GH * GWn;

  __shared__ float sA[64 * KPAD];    // GEMM A (wgLN out; later u*(1+v'))
  __shared__ float sBT[176 * KPAD];  // B transposed+padded (inW, then outW)
  __shared__ float sC[64 * LDC];     // GEMM C (162 cols pass1; 96 cols pass2)
  __shared__ float sS[64 * C_];      // shortcut (mw-LN output, 0 at pads)
  __shared__ float sGWm[F2 * F2];
  __shared__ float sGb[F2];
  __shared__ float sMg[C_], sMb[C_], sLg[C_], sLb[C_], sVg[C_], sVb[C_];
  __shared__ float sIb[P_], sOb[C_];

  const int tid = threadIdx.x;
  // phase 0: stage small params + B1 = inW (81 x 162) transposed into LDS
  for (int e = tid; e < F2*F2; e += 256) sGWm[e] = gW[e];
  for (int e = tid; e < F2;    e += 256) sGb[e]  = gb[e];
  for (int e = tid; e < C_;    e += 256) {
    sMg[e] = mwg[e]; sMb[e] = mwb[e];
    sLg[e] = lng[e]; sLb[e] = lnb[e];
    sVg[e] = slg[e]; sVb[e] = slb[e];
    sOb[e] = outb[e];
  }
  for (int e = tid; e < P_; e += 256) sIb[e] = inb[e];
  stage_BT(sBT, inW, P_, C_, P_, 176);
  __syncthreads();

  // phase 1: mw-LN (shortcut) then wg-LN into A
  if (tid < 64) {
    const int t = tid;
    sA[t*KPAD + 81] = 0.f; sA[t*KPAD + 82] = 0.f; sA[t*KPAD + 83] = 0.f;
    bool have = false;
    const float* xp = nullptr;
    if (t < M) {
      const int wl = t / F2, m = t % F2;
      const int wid = blockIdx.x * WPB + wl;
      if (wid < nWin) {
        const int n = wid / (GH * GWn);
        const int r = wid % (GH * GWn);
        const int gy = r / GWn, gx = r % GWn;
        const int dy = m / F, dx = m % F;
        const int h = gy*F + dy, x = gx*F + dx;
        if (h < H_ && x < W_) { have = true; xp = xin + (((size_t)n*H_ + h)*W_ + x) * C_; }
      }
    }
    if (have) {
      float mean = 0.f;
      for (int c = 0; c < C_; ++c) { float v = xp[c]; sS[t*C_ + c] = v; mean += v; }
      mean *= (1.f / C_);
      float var = 0.f;
      for (int c = 0; c < C_; ++c) { float d = sS[t*C_ + c] - mean; var += d*d; }
      var *= (1.f / C_);
      const float rs = rsqrtf(var + 1e-5f);
      for (int c = 0; c < C_; ++c)
        sS[t*C_ + c] = (sS[t*C_ + c] - mean) * rs * sMg[c] + sMb[c];
    } else {
      for (int c = 0; c < C_; ++c) sS[t*C_ + c] = 0.f;   // pad AFTER mw-LN => zeros
    }
    // wg-LN of shortcut row (LN(0) = beta, matching padded semantics)
    float mean = 0.f;
    for (int c = 0; c < C_; ++c) mean += sS[t*C_ + c];
    mean *= (1.f / C_);
    float var = 0.f;
    for (int c = 0; c < C_; ++c) { float d = sS[t*C_ + c] - mean; var += d*d; }
    var *= (1.f / C_);
    const float rs = rsqrtf(var + 1e-5f);
    if (t < M) {
      for (int c = 0; c < C_; ++c)
        sA[t*KPAD + c] = (sS[t*C_ + c] - mean) * rs * sLg[c] + sLb[c];
    } else {
      for (int c = 0; c < C_; ++c) sA[t*KPAD + c] = 0.f;  // zero filler rows
    }
  }
  __syncthreads();

  // phase 2: GEMM1 (tokens x 81) @ inW(81 x 162)
  wmma_gemm_lds(sA, sBT, sC, LDC, 11, P_);
  __syncthreads();

  // phase 3: bias + gelu, then LN over the v half; also restage sBT with outW
  if (tid < M) {
    const int t = tid;
    for (int c = 0; c < P_; ++c) sC[t*LDC + c] = gelu_exact(sC[t*LDC + c] + sIb[c]);
    float mean = 0.f;
    for (int c = 0; c < C_; ++c) mean += sC[t*LDC + 81 + c];
    mean *= (1.f / C_);
    float var = 0.f;
    for (int c = 0; c < C_; ++c) { float d = sC[t*LDC + 81 + c] - mean; var += d*d; }
    var *= (1.f / C_);
    const float rs = rsqrtf(var + 1e-5f);
    for (int c = 0; c < C_; ++c)
      sC[t*LDC + 81 + c] = (sC[t*LDC + 81 + c] - mean) * rs * sVg[c] + sVb[c];
  }
  stage_BT(sBT, outW, C_, C_, C_, 96);   // GEMM1 done with sBT; safe to overwrite
  __syncthreads();

  // phase 4: spatial gating per window + build A2 = u * (1 + sW@v + sb)
  for (int e = tid; e < M * C_; e += 256) {
    const int t = e / C_, c = e % C_;
    const int wl = t / F2, m = t % F2;
    float accv = sGb[m];
    for (int k = 0; k < F2; ++k)
      accv = fmaf(sGWm[m*F2 + k], sC[(wl*F2 + k)*LDC + 81 + c], accv);
    sA[t*KPAD + c] = sC[t*LDC + c] * (1.f + accv);
  }
  __syncthreads();

  // phase 5: GEMM2 (tokens x 81) @ outW(81 x 81)
  wmma_gemm_lds(sA, sBT, sC, LDC, 6, C_ + 15);  // cols 0..95 stored, only 0..80 used
  __syncthreads();

  // phase 6: + bias + shortcut, store valid tokens
  if (tid < M) {
    const int t = tid;
    const int wl = t / F2, m = t % F2;
    const int wid = blockIdx.x * WPB + wl;
    if (wid < nWin) {
      const int n = wid / (GH * GWn);
      const int r = wid % (GH * GWn);
      const int gy = r / GWn, gx = r % GWn;
      const int dy = m / F, dx = m % F;
      const int h = gy*F + dy, x = gx*F + dx;
      if (h < H_ && x < W_) {
        float* o = bout + (((size_t)n*H_ + h)*W_ + x) * C_;
        for (int c = 0; c < C_; ++c) o[c] = sC[t*LDC + c] + sOb[c] + sS[t*C_ + c];
      }
    }
  }
}

__global__ void zero_kernel(float* p, int n) {
  int i = blockIdx.x * blockDim.x + threadIdx.x;
  if (i < n) p[i] = 0.f;
}

// pooled sum of (b1+b2+b3) over (h,w) per (n,c)
__global__ void pool_kernel(const float* __restrict__ b1, const float* __restrict__ b2,
                            const float* __restrict__ b3, float* __restrict__ a) {
  const int n = blockIdx.y, h = blockIdx.x, c = threadIdx.x;
  if (c >= C_) return;
  float s = 0.f;
  const size_t base = (((size_t)n*H_ + h)*W_) * C_ + c;
  for (int w = 0; w < W_; ++w) {
    const size_t o = base + (size_t)w * C_;
    s += b1[o] + b2[o] + b3[o];
  }
  atomicAdd(&a[n*C_ + c], s);
}

// reweight MLP + softmax over 3 branches -> w0,w1,w2 (each 4x81)
__global__ __launch_bounds__(256)
void weights_kernel(const float* __restrict__ a,
                    const float* __restrict__ rw1W, const float* __restrict__ rw1b,
                    const float* __restrict__ rw2W, const float* __restrict__ rw2b,
                    float* __restrict__ wout) {
  __shared__ float am[4*C_];
  __shared__ float hm[4*20];
  __shared__ float wt[4*243];
  const int tid = threadIdx.x;
  for (int e = tid; e < 4*C_; e += 256) am[e] = a[e] * (1.f / (float)HW_);
  __syncthreads();
  for (int e = tid; e < 4*20; e += 256) {
    const int n = e / 20, j = e % 20;
    float s = rw1b[j];
    for (int c = 0; c < C_; ++c) s = fmaf(am[n*C_ + c], rw1W[c*20 + j], s);
    hm[e] = gelu_exact(s);
  }
  __syncthreads();
  for (int e = tid; e < 4*243; e += 256) {
    const int n = e / 243, q = e % 243;
    float s = rw2b[q];
    for (int j = 0; j < 20; ++j) s = fmaf(hm[n*20 + j], rw2W[j*243 + q], s);
    wt[e] = s;
  }
  __syncthreads();
  for (int e = tid; e < 4*C_; e += 256) {
    const int n = e / C_, c = e % C_;
    const float v0 = wt[n*243 + c*3], v1 = wt[n*243 + c*3 + 1], v2 = wt[n*243 + c*3 + 2];
    const float mx = fmaxf(v0, fmaxf(v1, v2));
    const float e0 = expf(v0 - mx), e1 = expf(v1 - mx), e2 = expf(v2 - mx);
    const float inv = 1.f / (e0 + e1 + e2);
    wout[0*4*C_ + n*C_ + c] = e0 * inv;
    wout[1*4*C_ + n*C_ + c] = e1 * inv;
    wout[2*4*C_ + n*C_ + c] = e2 * inv;
  }
}

// final: y = (b1*w0 + b2*w1 + b3*w2) @ out_W + out_b + x_in   (64 tokens / block)
__global__ __launch_bounds__(256)
void final_kernel(const float* __restrict__ b1, const float* __restrict__ b2,
                  const float* __restrict__ b3, const float* __restrict__ wts,
                  const float* __restrict__ outW, const float* __restrict__ outb,
                  const float* __restrict__ xin, float* __restrict__ out) {
  __shared__ float sA[64 * KPAD];
  __shared__ float sBT[96 * KPAD];
  __shared__ float sC[64 * 96];
  __shared__ float sw0[C_], sw1[C_], sw2[C_], sOb[C_];
  const int tid = threadIdx.x;
  const size_t base = (size_t)blockIdx.x * 64;
  const int n = (int)(base / HW_);   // 25600 % 64 == 0 -> block never spans images
  if (tid < C_) {
    sw0[tid] = wts[0*4*C_ + n*C_ + tid];
    sw1[tid] = wts[1*4*C_ + n*C_ + tid];
    sw2[tid] = wts[2*4*C_ + n*C_ + tid];
    sOb[tid] = outb[tid];
  }
  stage_BT(sBT, outW, C_, C_, C_, 96);
  __syncthreads();
  if (tid < 64) {
    const size_t o = (base + tid) * C_;
    for (int c = 0; c < C_; ++c)
      sA[tid*KPAD + c] = b1[o + c]*sw0[c] + b2[o + c]*sw1[c] + b3[o + c]*sw2[c];
    sA[tid*KPAD + 81] = 0.f; sA[tid*KPAD + 82] = 0.f; sA[tid*KPAD + 83] = 0.f;
  }
  __syncthreads();
  wmma_gemm_lds(sA, sBT, sC, 96, 6, 96);
  __syncthreads();
  if (tid < 64) {
    const size_t o = (base + tid) * C_;
    for (int c = 0; c < C_; ++c) out[o + c] = sC[tid*96 + c] + sOb[c] + xin[o + c];
  }
}

extern "C" void kernel_launch(void* const* d_in, const int* in_sizes, int n_in,
                              void* d_out, int out_size, void* d_ws, size_t ws_size,
                              hipStream_t stream) {
  const float* x1    = (const float*)d_in[0];
  const float* x2    = (const float*)d_in[1];
  const float* mwg   = (const float*)d_in[2];
  const float* mwb   = (const float*)d_in[3];
  const float* wlng  = (const float*)d_in[4];   // (3,81)
  const float* wlnb  = (const float*)d_in[5];
  const float* winW  = (const float*)d_in[6];   // (3,81,162)
  const float* winb  = (const float*)d_in[7];   // (3,162)
  const float* wslg  = (const float*)d_in[8];
  const float* wslb  = (const float*)d_in[9];
  const float* gW3   = (const float*)d_in[10];
  const float* gb3   = (const float*)d_in[11];
  const float* gW5   = (const float*)d_in[12];
  const float* gb5   = (const float*)d_in[13];
  const float* gW7   = (const float*)d_in[14];
  const float* gb7   = (const float*)d_in[15];
  const float* woutW = (const float*)d_in[16];  // (3,81,81)
  const float* woutb = (const float*)d_in[17];
  const float* rw1W  = (const float*)d_in[18];
  const float* rw1b  = (const float*)d_in[19];
  const float* rw2W  = (const float*)d_in[20];
  const float* rw2b  = (const float*)d_in[21];
  const float* outW  = (const float*)d_in[22];
  const float* outb  = (const float*)d_in[23];
  (void)in_sizes; (void)n_in; (void)out_size; (void)ws_size;

  float* ws  = (float*)d_ws;
  float* xin = ws;                              // 8,294,400
  float* b2  = xin + (size_t)NTOK_ * C_;        // 8,294,400
  float* b3  = b2  + (size_t)NTOK_ * C_;        // 8,294,400
  float* av  = b3  + (size_t)NTOK_ * C_;        // 324
  float* wts = av + 4*C_;                       // 972
  float* b1  = (float*)d_out;                   // reuse output buffer as b1

  // 1) correlation cost volume
  corr_kernel<<<dim3(100, 4), 256, 0, stream>>>(x1, x2, xin);

  // 2) three fused win_gmlp branches
  {
    const int gh3 = 54, nw3 = B_*gh3*gh3;                   // 11664 windows, 7/block
    branch_kernel<3, 7><<<(nw3 + 6) / 7, 256, 0, stream>>>(
        xin, mwg, mwb, wlng + 0*C_, wlnb + 0*C_, winW + 0*C_*P_, winb + 0*P_,
        wslg + 0*C_, wslb + 0*C_, gW3, gb3, woutW + 0*C_*C_, woutb + 0*C_, b1);
    const int gh5 = 32, nw5 = B_*gh5*gh5;                   // 4096 windows, 2/block
    branch_kernel<5, 2><<<(nw5 + 1) / 2, 256, 0, stream>>>(
        xin, mwg, mwb, wlng + 1*C_, wlnb + 1*C_, winW + 1*C_*P_, winb + 1*P_,
        wslg + 1*C_, wslb + 1*C_, gW5, gb5, woutW + 1*C_*C_, woutb + 1*C_, b2);
    const int gh7 = 23, nw7 = B_*gh7*gh7;                   // 2116 windows, 1/block
    branch_kernel<7, 1><<<nw7, 256, 0, stream>>>(
        xin, mwg, mwb, wlng + 2*C_, wlnb + 2*C_, winW + 2*C_*P_, winb + 2*P_,
        wslg + 2*C_, wslb + 2*C_, gW7, gb7, woutW + 2*C_*C_, woutb + 2*C_, b3);
  }

  // 3) pooled reweighting
  zero_kernel<<<(4*C_ + 255) / 256, 256, 0, stream>>>(av, 4*C_);
  pool_kernel<<<dim3(H_, B_), 96, 0, stream>>>(b1, b2, b3, av);
  weights_kernel<<<1, 256, 0, stream>>>(av, rw1W, rw1b, rw2W, rw2b, wts);

  // 4) weighted combine + output projection + residual (in-place safe per block)
  final_kernel<<<NTOK_ / 64, 256, 0, stream>>>(b1, b2, b3, wts, outW, outb, xin,
                                               (float*)d_out);
}